// GraAvg_9174050144814
// MI455X (gfx1250) — compile-verified
//
#include <hip/hip_runtime.h>
#include <hip/hip_bf16.h>

#define NN 100000
#define EE 1600000
#define DD 128
#define LL 3
#define PP 8

typedef __attribute__((ext_vector_type(2))) float v2f;
typedef __attribute__((ext_vector_type(8))) float v8f;

// ---------------------------------------------------------------------------
// Kernel 1: fold softmax(layer_weights[l]) into stacked_params -> ww[l][o][i]
// ---------------------------------------------------------------------------
__global__ __launch_bounds__(256) void build_ww_kernel(
    const float* __restrict__ sp,   // [L, D, D, P]
    const float* __restrict__ lw,   // [L, P]
    float* __restrict__ ww)         // [L, D, D]  row-major (o rows, i cols)
{
    int idx = blockIdx.x * blockDim.x + threadIdx.x;
    if (idx >= LL * DD * DD) return;
    int l = idx / (DD * DD);

    float v[PP];
    float m = -1e30f;
#pragma unroll
    for (int p = 0; p < PP; ++p) { v[p] = lw[l * PP + p]; m = fmaxf(m, v[p]); }
    float s = 0.f;
#pragma unroll
    for (int p = 0; p < PP; ++p) { v[p] = __expf(v[p] - m); s += v[p]; }
    float inv = 1.0f / s;

    const float* spb = sp + (size_t)idx * PP;   // idx == (l*D + o)*D + i
    float acc = 0.f;
#pragma unroll
    for (int p = 0; p < PP; ++p) acc = fmaf(spb[p], v[p], acc);
    ww[idx] = acc * inv;
}

// ---------------------------------------------------------------------------
// Kernel 2: zero-fill (float4 granularity)
// ---------------------------------------------------------------------------
__global__ __launch_bounds__(256) void zero_kernel(float4* __restrict__ p, int n4)
{
    int i = blockIdx.x * blockDim.x + threadIdx.x;
    if (i < n4) p[i] = make_float4(0.f, 0.f, 0.f, 0.f);
}

// ---------------------------------------------------------------------------
// Kernel 3: SpMM  ax = A @ h  with sorted edge_row segment-sum.
// One wave handles EDGES_PER_WAVE consecutive edges; lane owns 4 columns
// (float4 covering D=128 across 32 lanes). Row index is wave-uniform, so we
// accumulate in registers and flush with global f32 atomics on row change.
// ---------------------------------------------------------------------------
#define EDGES_PER_WAVE 512

__global__ __launch_bounds__(256) void spmm_kernel(
    const int*   __restrict__ erow,
    const int*   __restrict__ ecol,
    const float* __restrict__ eval,
    const float* __restrict__ h,    // [N, D]
    float*       __restrict__ ax)   // [N, D], pre-zeroed
{
    int wave = (blockIdx.x * blockDim.x + threadIdx.x) >> 5;
    int lane = threadIdx.x & 31;
    int e0 = wave * EDGES_PER_WAVE;
    if (e0 >= EE) return;
    int e1 = e0 + EDGES_PER_WAVE;
    if (e1 > EE) e1 = EE;

    const int c = lane * 4;
    float4 acc = make_float4(0.f, 0.f, 0.f, 0.f);
    int prev = erow[e0];

    for (int e = e0; e < e1; ++e) {
        int r = erow[e];                 // wave-uniform (lanes agree)
        if (r != prev) {
            float* dst = ax + (size_t)prev * DD + c;
            atomicAdd(dst + 0, acc.x);
            atomicAdd(dst + 1, acc.y);
            atomicAdd(dst + 2, acc.z);
            atomicAdd(dst + 3, acc.w);
            acc = make_float4(0.f, 0.f, 0.f, 0.f);
            prev = r;
        }
        int   col = ecol[e];
        float val = eval[e];
        const float4 hv = *(const float4*)(h + (size_t)col * DD + c);
        acc.x = fmaf(hv.x, val, acc.x);
        acc.y = fmaf(hv.y, val, acc.y);
        acc.z = fmaf(hv.z, val, acc.z);
        acc.w = fmaf(hv.w, val, acc.w);
    }
    float* dst = ax + (size_t)prev * DD + c;
    atomicAdd(dst + 0, acc.x);
    atomicAdd(dst + 1, acc.y);
    atomicAdd(dst + 2, acc.z);
    atomicAdd(dst + 3, acc.w);
}

// ---------------------------------------------------------------------------
// Kernel 4: h_out[n][o] = (relu of) sum_i ax[n][i] * ww[o][i]
// FP32 WMMA (v_wmma_f32_16x16x4_f32). Block = 256 threads (8 waves),
// computes a 32(M) x 128(N) output tile, K tiled by 64 through LDS.
// LDS pitch 68 floats (pitch%64 == 4) -> bank-conflict-free v2f fragment
// reads. Wave w owns column-group w and two 16x16 accumulators (row groups
// 0 and 1); the B fragment is loaded once per k-step and reused for both.
// ---------------------------------------------------------------------------
#define PITCH 68
#define KTILE 64

__global__ __launch_bounds__(256) void gemm_wmma_kernel(
    const float* __restrict__ ax,   // [N, D]
    const float* __restrict__ ww,   // [D, D]  (o rows, i cols)
    float*       __restrict__ out,  // [N, D]
    int do_relu)
{
    __shared__ __align__(16) float sA[32 * PITCH];    // ax tile  [32][64]
    __shared__ __align__(16) float sB[DD * PITCH];    // ww tile  [128][64]

    const int tid     = threadIdx.x;
    const int lane    = tid & 31;
    const int cg      = tid >> 5;          // wave id == column group 0..7
    const int nl      = lane & 15;
    const int half    = lane >> 4;
    const int rowBase = blockIdx.x * 32;

    v8f acc0 = {};   // rows [rowBase+0 .. +15]  x cols [cg*16 .. +15]
    v8f acc1 = {};   // rows [rowBase+16 .. +31] x cols [cg*16 .. +15]

    for (int kt = 0; kt < DD; kt += KTILE) {
        // --- stage A tile: 32 rows x 64 cols, coalesced float4 loads -------
#pragma unroll
        for (int i = 0; i < 2; ++i) {
            int idx = tid + i * 256;               // 0..511
            int r   = idx >> 4;
            int c4  = (idx & 15) * 4;
            int row = rowBase + r;
            if (row >= NN) row = NN - 1;           // clamp (results unused)
            float4 v = *(const float4*)(ax + (size_t)row * DD + kt + c4);
            float* d = &sA[r * PITCH + c4];
            d[0] = v.x; d[1] = v.y; d[2] = v.z; d[3] = v.w;
        }
        // --- stage B tile: 128 rows (o) x 64 cols (i) ----------------------
#pragma unroll
        for (int i = 0; i < 8; ++i) {
            int idx = tid + i * 256;               // 0..2047
            int o   = idx >> 4;
            int c4  = (idx & 15) * 4;
            float4 v = *(const float4*)(ww + (size_t)o * DD + kt + c4);
            float* d = &sB[o * PITCH + c4];
            d[0] = v.x; d[1] = v.y; d[2] = v.z; d[3] = v.w;
        }
        __syncthreads();

        // --- 16 k-steps of v_wmma_f32_16x16x4_f32 --------------------------
#pragma unroll
        for (int ks = 0; ks < KTILE / 4; ++ks) {
            const int koff = ks * 4 + 2 * half;    // lane's two K values
            v2f b  = *(const v2f*)&sB[(cg * 16 + nl) * PITCH + koff];
            v2f a0 = *(const v2f*)&sA[(nl)           * PITCH + koff];
            v2f a1 = *(const v2f*)&sA[(16 + nl)      * PITCH + koff];
            acc0 = __builtin_amdgcn_wmma_f32_16x16x4_f32(
                false, a0, false, b, (short)0, acc0, false, false);
            acc1 = __builtin_amdgcn_wmma_f32_16x16x4_f32(
                false, a1, false, b, (short)0, acc1, false, false);
        }
        __syncthreads();
    }

    // --- store C: VGPR r holds M = r + 8*half, N = nl -----------------------
    const int col = cg * 16 + nl;
#pragma unroll
    for (int r = 0; r < 8; ++r) {
        int row0 = rowBase + r + 8 * half;
        int row1 = row0 + 16;
        float v0 = acc0[r];
        float v1 = acc1[r];
        if (do_relu) { v0 = fmaxf(v0, 0.f); v1 = fmaxf(v1, 0.f); }
        if (row0 < NN) out[(size_t)row0 * DD + col] = v0;
        if (row1 < NN) out[(size_t)row1 * DD + col] = v1;
    }
}

// ---------------------------------------------------------------------------
extern "C" void kernel_launch(void* const* d_in, const int* in_sizes, int n_in,
                              void* d_out, int out_size, void* d_ws, size_t ws_size,
                              hipStream_t stream) {
    (void)in_sizes; (void)n_in; (void)out_size; (void)ws_size;

    const int*   erow = (const int*)  d_in[0];
    const int*   ecol = (const int*)  d_in[1];
    const float* eval = (const float*)d_in[2];
    const float* x    = (const float*)d_in[3];
    const float* sp   = (const float*)d_in[4];
    const float* lw   = (const float*)d_in[5];
    float* out = (float*)d_out;

    float* ww   = (float*)d_ws;                  // L*D*D floats
    float* ax   = ww + (size_t)LL * DD * DD;     // N*D floats
    float* hbuf = ax + (size_t)NN * DD;          // N*D floats

    // 0) combined weights for all layers
    build_ww_kernel<<<(LL * DD * DD + 255) / 256, 256, 0, stream>>>(sp, lw, ww);

    const int n4         = NN * DD / 4;
    const int zero_grid  = (n4 + 255) / 256;
    const int spmm_waves = (EE + EDGES_PER_WAVE - 1) / EDGES_PER_WAVE;
    const int spmm_grid  = (spmm_waves + 7) / 8;          // 8 waves / block
    const int gemm_grid  = (NN + 31) / 32;

    for (int layer = 0; layer < LL; ++layer) {
        const float* hin  = (layer == 0) ? x : hbuf;
        float*       hout = (layer == LL - 1) ? out : hbuf;

        zero_kernel<<<zero_grid, 256, 0, stream>>>((float4*)ax, n4);
        spmm_kernel<<<spmm_grid, 256, 0, stream>>>(erow, ecol, eval, hin, ax);
        gemm_wmma_kernel<<<gemm_grid, 256, 0, stream>>>(
            ax, ww + (size_t)layer * DD * DD, hout, (layer < LL - 1) ? 1 : 0);
    }
}